// MultiHeadSSM_34961033790197
// MI455X (gfx1250) — compile-verified
//
#include <hip/hip_runtime.h>
#include <hip/hip_bf16.h>

// ---------------------------------------------------------------------------
// MultiHeadSSM for MI455X (gfx1250, wave32).
//  proj_kernel    : xp = X(16x64-tile) @ Wx^T (WMMA f32 16x16x4), dt GEMM + softplus
//  scan_local     : chunked SSM scan, h0=0 per chunk (64 chunks of 32 steps)
//  scan_combine   : serial chunk-carry combine via exp(A * sum(dt)) identity
//  scan_fixup     : add carried-state correction to y
//  outgemm_kernel : y(4096x1024) @ Wout^T + bout (WMMA f32 16x16x4)
// ---------------------------------------------------------------------------

typedef __attribute__((ext_vector_type(2))) float v2f;
typedef __attribute__((ext_vector_type(4))) float v4f;
typedef __attribute__((ext_vector_type(8))) float v8f;

#define D_MODEL 1024
#define D_STATE 16
#define N_HEADS 16
#define DT_RANK 64
#define D_HEAD  64
#define BATCH   2
#define SEQ     2048
#define NROWS   (BATCH * SEQ * N_HEADS)   // 65536 (b,l,h) rows
#define NP      96                        // DT_RANK + 2*D_STATE
#define NCHUNK  64                        // chunks per sequence
#define CLEN    (SEQ / NCHUNK)            // 32 steps per chunk

// D = A(16x4) * B(4x16) + C, fp32. Lane l: M/N = l%16, K-pair = 2*(l/16)+{v0,v1}.
__device__ __forceinline__ v8f wmma_k4(v2f a, v2f b, v8f c) {
  return __builtin_amdgcn_wmma_f32_16x16x4_f32(
      /*neg_a=*/false, a, /*neg_b=*/false, b,
      /*c_mod=*/(short)0, c, /*reuse_a=*/false, /*reuse_b=*/false);
}

__device__ __forceinline__ float softplusf(float z) {
  return (z > 20.0f) ? z : log1pf(__expf(z));
}

// ---------------------------------------------------------------------------
// Kernel 1: per 16-row tile (rows = flattened (b,l,h); X row = xh[b,l,h,:]):
//   xp(16x96) = X(16x64) @ Wx^T        -> staged in LDS
//   dt(16x64) = softplus(xp[:, :64] @ Wdt^T + bdt)
//   B_x, C_x  = xp[:, 64:80], xp[:, 80:96]
// 4 independent waves per 128-thread block, one tile each.
// ---------------------------------------------------------------------------
__global__ __launch_bounds__(128) void proj_kernel(
    const float* __restrict__ x,   const float* __restrict__ Wx,
    const float* __restrict__ Wdt, const float* __restrict__ bdt,
    float* __restrict__ dt_ws, float* __restrict__ bx_ws, float* __restrict__ cx_ws)
{
  __shared__ float lds[4][16 * NP];

  const int wid   = threadIdx.x >> 5;
  const int lane  = threadIdx.x & 31;
  const int mrow  = lane & 15;   // M (or N) index within tile
  const int khalf = lane >> 4;   // selects K-pair {0,1} vs {2,3}
  const int row0  = (blockIdx.x * 4 + wid) * 16;

  float* ltile = lds[wid];

  // ---- GEMM1: preload A tile (16 k-steps of float2 per lane) ----
  const float* aptr = x + (size_t)(row0 + mrow) * D_HEAD + 2 * khalf;
  v2f areg[16];
#pragma unroll
  for (int kk = 0; kk < 16; ++kk)
    areg[kk] = *(const v2f*)(aptr + kk * 4);

#pragma unroll
  for (int nt = 0; nt < 6; ++nt) {                 // 96 = 6 * 16 output cols
    const float* bptr = Wx + (size_t)(nt * 16 + mrow) * D_HEAD + 2 * khalf;
    v8f acc = {};
#pragma unroll
    for (int kk = 0; kk < 16; ++kk) {
      v2f bv = *(const v2f*)(bptr + kk * 4);
      acc = wmma_k4(areg[kk], bv, acc);
    }
    // C layout: vgpr r -> row (r + 8*khalf), col = nt*16 + mrow
#pragma unroll
    for (int r = 0; r < 8; ++r)
      ltile[(r + 8 * khalf) * NP + nt * 16 + mrow] = acc[r];
  }
  __syncthreads();

  // ---- GEMM2: dt_pre(16x64) = xp[:, :64] @ Wdt^T ----
  v8f acc2[4];
#pragma unroll
  for (int j = 0; j < 4; ++j) { v8f z = {}; acc2[j] = z; }

#pragma unroll
  for (int kk = 0; kk < 16; ++kk) {
    v2f a2 = *(const v2f*)(ltile + mrow * NP + kk * 4 + 2 * khalf);
#pragma unroll
    for (int j = 0; j < 4; ++j) {
      v2f bv = *(const v2f*)(Wdt + (size_t)(j * 16 + mrow) * DT_RANK + kk * 4 + 2 * khalf);
      acc2[j] = wmma_k4(a2, bv, acc2[j]);
    }
  }

  // softplus(z + bdt) -> dt_ws[(row)*64 + col]
#pragma unroll
  for (int j = 0; j < 4; ++j) {
    const int col  = j * 16 + mrow;
    const float bz = bdt[col];
#pragma unroll
    for (int r = 0; r < 8; ++r) {
      const int m = r + 8 * khalf;
      dt_ws[(size_t)(row0 + m) * D_HEAD + col] = softplusf(acc2[j][r] + bz);
    }
  }

  // ---- scatter B_x / C_x (cols 64..79 / 80..95). khalf picks which matrix. ----
  {
    float* dst = (khalf == 0) ? bx_ws : cx_ws;
    const int coff = 64 + 16 * khalf;
#pragma unroll
    for (int q = 0; q < 4; ++q) {
      v4f v = *(const v4f*)(ltile + mrow * NP + coff + q * 4);
      *(v4f*)(dst + (size_t)(row0 + mrow) * D_STATE + q * 4) = v;
    }
  }
}

// ---------------------------------------------------------------------------
// Kernel 2a: chunk-local scan with h0 = 0. One block per (b,h,chunk);
// thread d owns state h[d, 0:16] in registers. Also emits the chunk's final
// local state hL and sum(dt) (the chunk decay is exp(A_n * sum dt)).
// ---------------------------------------------------------------------------
__global__ __launch_bounds__(64) void scan_local(
    const float* __restrict__ x,     const float* __restrict__ A_log,
    const float* __restrict__ dt_ws, const float* __restrict__ bx_ws,
    const float* __restrict__ cx_ws, float* __restrict__ y_ws,
    float* __restrict__ hL_ws, float* __restrict__ S_ws)
{
  const int bh = blockIdx.x / NCHUNK;      // (b*16 + h)
  const int c  = blockIdx.x % NCHUNK;
  const int b  = bh >> 4;
  const int h  = bh & 15;
  const int d  = threadIdx.x;

  float An[D_STATE];
#pragma unroll
  for (int n = 0; n < D_STATE; ++n)
    An[n] = -__expf(A_log[h * D_STATE + n]);

  float s[D_STATE];
#pragma unroll
  for (int n = 0; n < D_STATE; ++n) s[n] = 0.0f;
  float Ssum = 0.0f;

  for (int tt = 0; tt < CLEN; ++tt) {
    const int t      = c * CLEN + tt;
    const size_t r   = (size_t)(b * SEQ + t) * N_HEADS + h;
    const float dtv  = dt_ws[r * D_HEAD + d];
    const float xv   = x[(size_t)(b * SEQ + t) * D_MODEL + h * D_HEAD + d];
    const float dx   = dtv * xv;
    float y = 0.0f;
#pragma unroll
    for (int n = 0; n < D_STATE; ++n) {
      const float Bn = bx_ws[r * D_STATE + n];
      const float Cn = cx_ws[r * D_STATE + n];
      const float ab = __expf(dtv * An[n]);
      s[n] = fmaf(ab, s[n], dx * Bn);
      y    = fmaf(s[n], Cn, y);
    }
    y_ws[(size_t)(b * SEQ + t) * D_MODEL + h * D_HEAD + d] = y;
    Ssum += dtv;
  }

  float* hdst = hL_ws + ((size_t)(bh * NCHUNK + c) * D_HEAD + d) * D_STATE;
#pragma unroll
  for (int q = 0; q < 4; ++q) {
    v4f v = { s[q * 4 + 0], s[q * 4 + 1], s[q * 4 + 2], s[q * 4 + 3] };
    *(v4f*)(hdst + q * 4) = v;
  }
  S_ws[(size_t)(bh * NCHUNK + c) * D_HEAD + d] = Ssum;
}

// ---------------------------------------------------------------------------
// Kernel 2b: serial carry across chunks (in-place: hL -> h_in).
//   h_in[c] = exp(A * S[c-1]) * h_in[c-1] + hL[c-1],  h_in[0] = 0
// One block per (b,h); thread d sweeps its 64 chunks (serial depth 64).
// ---------------------------------------------------------------------------
__global__ __launch_bounds__(64) void scan_combine(
    const float* __restrict__ A_log,
    float* __restrict__ hL_ws, const float* __restrict__ S_ws)
{
  const int bh = blockIdx.x;
  const int h  = bh & 15;
  const int d  = threadIdx.x;

  float An[D_STATE];
#pragma unroll
  for (int n = 0; n < D_STATE; ++n)
    An[n] = -__expf(A_log[h * D_STATE + n]);

  float carry[D_STATE];
#pragma unroll
  for (int n = 0; n < D_STATE; ++n) carry[n] = 0.0f;

  for (int c = 0; c < NCHUNK; ++c) {
    float* hptr = hL_ws + ((size_t)(bh * NCHUNK + c) * D_HEAD + d) * D_STATE;
    const float Sc = S_ws[(size_t)(bh * NCHUNK + c) * D_HEAD + d];
    float tmp[D_STATE];
#pragma unroll
    for (int q = 0; q < 4; ++q) {
      v4f v = *(const v4f*)(hptr + q * 4);
      tmp[q * 4 + 0] = v[0]; tmp[q * 4 + 1] = v[1];
      tmp[q * 4 + 2] = v[2]; tmp[q * 4 + 3] = v[3];
    }
    // write the state ENTERING chunk c, then advance the carry past chunk c
#pragma unroll
    for (int q = 0; q < 4; ++q) {
      v4f v = { carry[q * 4 + 0], carry[q * 4 + 1], carry[q * 4 + 2], carry[q * 4 + 3] };
      *(v4f*)(hptr + q * 4) = v;
    }
#pragma unroll
    for (int n = 0; n < D_STATE; ++n)
      carry[n] = fmaf(__expf(An[n] * Sc), carry[n], tmp[n]);
  }
}

// ---------------------------------------------------------------------------
// Kernel 2c: correction pass. For chunk c with incoming state h_in:
//   y_t += sum_n (prod_{s<=t} exp(dt_s*A_n)) * h_in[n] * C_t[n]
// Track r[n] = h_in[n] and decay it by exp(dt_t*A_n) each step.
// ---------------------------------------------------------------------------
__global__ __launch_bounds__(64) void scan_fixup(
    const float* __restrict__ A_log,
    const float* __restrict__ dt_ws, const float* __restrict__ cx_ws,
    const float* __restrict__ hL_ws,   // now holds h_in
    float* __restrict__ y_ws)
{
  const int bh = blockIdx.x / NCHUNK;
  const int c  = blockIdx.x % NCHUNK;
  const int b  = bh >> 4;
  const int h  = bh & 15;
  const int d  = threadIdx.x;

  float An[D_STATE];
#pragma unroll
  for (int n = 0; n < D_STATE; ++n)
    An[n] = -__expf(A_log[h * D_STATE + n]);

  float rdec[D_STATE];
  const float* hsrc = hL_ws + ((size_t)(bh * NCHUNK + c) * D_HEAD + d) * D_STATE;
#pragma unroll
  for (int q = 0; q < 4; ++q) {
    v4f v = *(const v4f*)(hsrc + q * 4);
    rdec[q * 4 + 0] = v[0]; rdec[q * 4 + 1] = v[1];
    rdec[q * 4 + 2] = v[2]; rdec[q * 4 + 3] = v[3];
  }

  for (int tt = 0; tt < CLEN; ++tt) {
    const int t      = c * CLEN + tt;
    const size_t r   = (size_t)(b * SEQ + t) * N_HEADS + h;
    const float dtv  = dt_ws[r * D_HEAD + d];
    float ycorr = 0.0f;
#pragma unroll
    for (int n = 0; n < D_STATE; ++n) {
      rdec[n] *= __expf(dtv * An[n]);
      ycorr = fmaf(rdec[n], cx_ws[r * D_STATE + n], ycorr);
    }
    y_ws[(size_t)(b * SEQ + t) * D_MODEL + h * D_HEAD + d] += ycorr;
  }
}

// ---------------------------------------------------------------------------
// Kernel 3: out = y(4096x1024) @ Wout^T + bout.
// One wave -> 16x64 output tile (A reused across 4 N-subtiles), K = 256 steps.
// Everything lives in the 192MB L2, so the K-stream stays hot.
// ---------------------------------------------------------------------------
__global__ __launch_bounds__(128) void outgemm_kernel(
    const float* __restrict__ y_ws, const float* __restrict__ Wout,
    const float* __restrict__ bout, float* __restrict__ out)
{
  const int wid   = threadIdx.x >> 5;
  const int lane  = threadIdx.x & 31;
  const int mrow  = lane & 15;
  const int khalf = lane >> 4;
  const int gw    = blockIdx.x * 4 + wid;   // 4096 waves total
  const int m0    = (gw >> 4) * 16;         // 256 M tiles
  const int n0    = (gw & 15) * 64;         // 16 N groups of 64

  const float* aptr  = y_ws + (size_t)(m0 + mrow) * D_MODEL + 2 * khalf;
  const float* bptr0 = Wout + (size_t)(n0 + 0 * 16 + mrow) * D_MODEL + 2 * khalf;
  const float* bptr1 = Wout + (size_t)(n0 + 1 * 16 + mrow) * D_MODEL + 2 * khalf;
  const float* bptr2 = Wout + (size_t)(n0 + 2 * 16 + mrow) * D_MODEL + 2 * khalf;
  const float* bptr3 = Wout + (size_t)(n0 + 3 * 16 + mrow) * D_MODEL + 2 * khalf;

  v8f acc0 = {}, acc1 = {}, acc2 = {}, acc3 = {};

#pragma unroll 4
  for (int kk = 0; kk < 256; ++kk) {
    if ((kk & 31) == 0 && kk < 224)
      __builtin_prefetch(aptr + (kk + 32) * 4, 0, 3);   // global_prefetch on A stream
    v2f a  = *(const v2f*)(aptr  + kk * 4);
    v2f b0 = *(const v2f*)(bptr0 + kk * 4);
    v2f b1 = *(const v2f*)(bptr1 + kk * 4);
    v2f b2 = *(const v2f*)(bptr2 + kk * 4);
    v2f b3 = *(const v2f*)(bptr3 + kk * 4);
    acc0 = wmma_k4(a, b0, acc0);
    acc1 = wmma_k4(a, b1, acc1);
    acc2 = wmma_k4(a, b2, acc2);
    acc3 = wmma_k4(a, b3, acc3);
  }

  v8f accs[4] = {acc0, acc1, acc2, acc3};
#pragma unroll
  for (int j = 0; j < 4; ++j) {
    const int col  = n0 + j * 16 + mrow;
    const float bz = bout[col];
#pragma unroll
    for (int r = 0; r < 8; ++r) {
      const int m = m0 + r + 8 * khalf;
      out[(size_t)m * D_MODEL + col] = accs[j][r] + bz;
    }
  }
}

// ---------------------------------------------------------------------------
extern "C" void kernel_launch(void* const* d_in, const int* in_sizes, int n_in,
                              void* d_out, int out_size, void* d_ws, size_t ws_size,
                              hipStream_t stream) {
  const float* x     = (const float*)d_in[0];
  const float* A_log = (const float*)d_in[1];
  const float* Wx    = (const float*)d_in[2];
  const float* Wdt   = (const float*)d_in[3];
  const float* bdt   = (const float*)d_in[4];
  const float* Wout  = (const float*)d_in[5];
  const float* bout  = (const float*)d_in[6];
  float* out = (float*)d_out;

  float* ws    = (float*)d_ws;
  float* dt_ws = ws;                                            // 65536*64   = 16 MB
  float* bx_ws = dt_ws + (size_t)NROWS * D_HEAD;                // 65536*16   =  4 MB
  float* cx_ws = bx_ws + (size_t)NROWS * D_STATE;               // 65536*16   =  4 MB
  float* y_ws  = cx_ws + (size_t)NROWS * D_STATE;               // 4096*1024  = 16 MB
  float* hL_ws = y_ws  + (size_t)BATCH * SEQ * D_MODEL;         // 32*64*64*16 = 8 MB
  float* S_ws  = hL_ws + (size_t)BATCH * N_HEADS * NCHUNK * D_HEAD * D_STATE; // 0.5 MB

  // Phase 1: projections (WMMA).  4096 tiles, 4 waves/block.
  proj_kernel<<<NROWS / 16 / 4, 128, 0, stream>>>(x, Wx, Wdt, bdt, dt_ws, bx_ws, cx_ws);

  // Phase 2: chunked scan. 2a: local scans (2048 blocks), 2b: carry (32 blocks),
  //          2c: correction (2048 blocks). Serial depth 32+64+32 vs 2048.
  scan_local<<<BATCH * N_HEADS * NCHUNK, 64, 0, stream>>>(
      x, A_log, dt_ws, bx_ws, cx_ws, y_ws, hL_ws, S_ws);
  scan_combine<<<BATCH * N_HEADS, 64, 0, stream>>>(A_log, hL_ws, S_ws);
  scan_fixup<<<BATCH * N_HEADS * NCHUNK, 64, 0, stream>>>(
      A_log, dt_ws, cx_ws, hL_ws, y_ws);

  // Phase 3: output GEMM (WMMA).  4096 waves, 4 per block.
  outgemm_kernel<<<(BATCH * SEQ / 16) * (D_MODEL / 64) / 4, 128, 0, stream>>>(y_ws, Wout, bout, out);
}